// LocationAttention_14405320311333
// MI455X (gfx1250) — compile-verified
//
#include <hip/hip_runtime.h>

// ---------------- problem constants ----------------
constexpr int kB = 64, kT = 4096, kDIM = 256, kNLOC = 32, kKS = 31, kPAD = 15;
constexpr int kNS = 8;                  // T-chunks per batch row
constexpr int kTCHUNK = kT / kNS;       // 512 rows per workgroup
constexpr int kROWS = kTCHUNK / 8;      // 64 rows per wave (8 waves / block)

// ---------------- workspace layout (float offsets) ----------------
constexpr size_t WQ_OFF   = 0;                                   // B*DIM
constexpr size_t CW_OFF   = WQ_OFF   + (size_t)kB * kDIM;        // B*32 (KS padded)
constexpr size_t CB_OFF   = CW_OFF   + (size_t)kB * 32;          // B
constexpr size_t LOCE_OFF = CB_OFF   + 64;                       // B*T
constexpr size_t E_OFF    = LOCE_OFF + (size_t)kB * kT;          // B*T
constexpr size_t PM_OFF   = E_OFF    + (size_t)kB * kT;          // B*NS
constexpr size_t PS_OFF   = PM_OFF   + (size_t)kB * kNS;         // B*NS
constexpr size_t PC_OFF   = PS_OFF   + (size_t)kB * kNS;         // B*NS*DIM
constexpr size_t MS_OFF   = PC_OFF   + (size_t)kB * kNS * kDIM;  // 2*B

typedef float v2f __attribute__((ext_vector_type(2)));
typedef float v8f __attribute__((ext_vector_type(8)));

// ============ kernel 1: lq = q@L_w ; cw = lq@conv_w ; cb = lq@conv_b ============
__global__ void k_prep(const float* __restrict__ q, const float* __restrict__ Lw,
                       const float* __restrict__ convw, const float* __restrict__ convb,
                       float* __restrict__ ws) {
  __shared__ float lq[kB * kNLOC];
  const int tid = threadIdx.x;
  for (int i = tid; i < kB * kNLOC; i += 256) {
    const int b = i >> 5, c = i & 31;
    float acc = 0.f;
    for (int e = 0; e < kDIM; ++e) acc += q[b * kDIM + e] * Lw[e * kNLOC + c];
    lq[i] = acc;
  }
  __syncthreads();
  for (int i = tid; i < kB * kKS; i += 256) {
    const int b = i / kKS, k = i % kKS;
    float acc = 0.f;
    for (int c = 0; c < kNLOC; ++c) acc += lq[b * kNLOC + c] * convw[c * kKS + k];
    ws[CW_OFF + b * 32 + k] = acc;
  }
  if (tid < kB) {
    float acc = 0.f;
    for (int c = 0; c < kNLOC; ++c) acc += lq[tid * kNLOC + c] * convb[c];
    ws[CB_OFF + tid] = acc;
  }
}

// ============ kernel 2: wq[b,d] = sum_e q[b,e]*W_w[e,d] via V_WMMA_F32_16X16X4_F32 ====
// One wave per 16x16 output tile; 64 tiles (4 b-tiles x 16 d-tiles); K=256 in steps of 4.
// A 16x4 layout: lanes0-15 M=0..15 {v0:K0, v1:K1}; lanes16-31 {v0:K2, v1:K3}.
// B 4x16 layout mirrors A; D 16x16: vgpr r = row M=r (lanes<16) / M=r+8 (lanes>=16).
__global__ void k_wq(const float* __restrict__ q, const float* __restrict__ W,
                     float* __restrict__ wq) {
  const int tile = blockIdx.x;
  const int b0 = (tile >> 4) * 16;
  const int d0 = (tile & 15) * 16;
  const int l = threadIdx.x;
  const int half = l >> 4;      // 0: K rows {0,1}; 1: K rows {2,3}
  const int lm = l & 15;
  v8f acc = {};
  for (int e0 = 0; e0 < kDIM; e0 += 4) {
    v2f a, bm;
    const float* qp = q + (b0 + lm) * kDIM + e0 + 2 * half;
    a.x = qp[0]; a.y = qp[1];
    const float* wp = W + (e0 + 2 * half) * kDIM + d0 + lm;
    bm.x = wp[0]; bm.y = wp[kDIM];
    acc = __builtin_amdgcn_wmma_f32_16x16x4_f32(false, a, false, bm,
                                                (short)0, acc, false, false);
  }
#pragma unroll
  for (int r = 0; r < 8; ++r)
    wq[(b0 + r + 8 * half) * kDIM + d0 + lm] = acc[r];
}

// ============ kernel 3: locE[b,t] = cb[b] + sum_k cw[b,k]*mask[b,t+k-PAD] ============
__global__ void k_loce(const float* __restrict__ mask, float* __restrict__ ws) {
  const int i = blockIdx.x * 256 + threadIdx.x;   // i = b*T + t
  const int b = i >> 12, t = i & (kT - 1);
  float acc = ws[CB_OFF + b];
#pragma unroll
  for (int k = 0; k < kKS; ++k) {
    const int tt = t + k - kPAD;
    if (tt >= 0 && tt < kT) acc += ws[CW_OFF + b * 32 + k] * mask[b * kT + tt];
  }
  ws[LOCE_OFF + i] = acc;
}

// ============ kernel 4: fused energies + online softmax + context accumulation =======
// block = (chunk, b); 8 waves, each streams 64 encoder rows (1 KB each, coalesced
// float4), computes e = row.wq + locE, keeps (m, s, c[]) flash-style accumulators.
__global__ __launch_bounds__(256) void k_main(const float* __restrict__ enc,
                                              float* __restrict__ ws) {
  const int chunk = blockIdx.x;
  const int b = blockIdx.y;
  const int tid = threadIdx.x, w = tid >> 5, lane = tid & 31;
  __shared__ float s_m[8], s_s[8];
  __shared__ __align__(16) float s_c[8 * kDIM];

  const float4* wq4 = (const float4*)(ws + WQ_OFF + (size_t)b * kDIM);
  const float4 wa = wq4[lane];
  const float4 wb = wq4[32 + lane];
  const float* locE = ws + LOCE_OFF + (size_t)b * kT;
  float* E = ws + E_OFF + (size_t)b * kT;

  float m = -3.0e38f, s = 0.f;
  float4 accA = {0.f, 0.f, 0.f, 0.f}, accB = {0.f, 0.f, 0.f, 0.f};
  const int t0 = chunk * kTCHUNK + w * kROWS;

  for (int r = 0; r < kROWS; ++r) {
    const int t = t0 + r;
    const float* rowp = enc + ((size_t)t * kB + b) * kDIM;
    if (r + 2 < kROWS) {
      const float* nrow = enc + ((size_t)(t + 2) * kB + b) * kDIM;
      __builtin_prefetch(nrow + lane * 4, 0, 0);
      __builtin_prefetch(nrow + 128 + lane * 4, 0, 0);
    }
    const float4 a  = ((const float4*)rowp)[lane];
    const float4 bv = ((const float4*)rowp)[32 + lane];
    float e = a.x * wa.x + a.y * wa.y + a.z * wa.z + a.w * wa.w
            + bv.x * wb.x + bv.y * wb.y + bv.z * wb.z + bv.w * wb.w;
    e += __shfl_xor(e, 16, 32);
    e += __shfl_xor(e, 8, 32);
    e += __shfl_xor(e, 4, 32);
    e += __shfl_xor(e, 2, 32);
    e += __shfl_xor(e, 1, 32);
    e += locE[t];
    if (lane == 0) E[t] = e;
    const float mN  = fmaxf(m, e);
    const float scl = __expf(m - mN);   // 0 on first iter (m = -big)
    const float p   = __expf(e - mN);
    s = s * scl + p;
    accA.x = accA.x * scl + p * a.x;  accA.y = accA.y * scl + p * a.y;
    accA.z = accA.z * scl + p * a.z;  accA.w = accA.w * scl + p * a.w;
    accB.x = accB.x * scl + p * bv.x; accB.y = accB.y * scl + p * bv.y;
    accB.z = accB.z * scl + p * bv.z; accB.w = accB.w * scl + p * bv.w;
    m = mN;
  }

  // merge the 8 waves of this block in LDS
  ((float4*)(s_c + w * kDIM))[lane] = accA;
  ((float4*)(s_c + w * kDIM + 128))[lane] = accB;
  if (lane == 0) { s_m[w] = m; s_s[w] = s; }
  __syncthreads();

  const int d = tid;                    // 256 threads == kDIM
  float mwg = s_m[0];
#pragma unroll
  for (int i = 1; i < 8; ++i) mwg = fmaxf(mwg, s_m[i]);
  float ctx = 0.f, swg = 0.f;
#pragma unroll
  for (int i = 0; i < 8; ++i) {
    const float f = __expf(s_m[i] - mwg);
    swg += s_s[i] * f;
    ctx += s_c[i * kDIM + d] * f;
  }
  const size_t pidx = (size_t)(b * kNS + chunk);
  ws[PC_OFF + pidx * kDIM + d] = ctx;
  if (tid == 0) { ws[PM_OFF + pidx] = mwg; ws[PS_OFF + pidx] = swg; }
}

// ============ kernel 5: merge chunk partials; emit context; save (m_b, s_b) ==========
__global__ void k_combine(float* __restrict__ ws, float* __restrict__ out) {
  const int b = blockIdx.x, d = threadIdx.x;
  float mb = ws[PM_OFF + b * kNS];
  for (int i = 1; i < kNS; ++i) mb = fmaxf(mb, ws[PM_OFF + b * kNS + i]);
  float sb = 0.f, ctx = 0.f;
  for (int i = 0; i < kNS; ++i) {
    const float f = __expf(ws[PM_OFF + b * kNS + i] - mb);
    sb  += ws[PS_OFF + b * kNS + i] * f;
    ctx += ws[PC_OFF + (size_t)(b * kNS + i) * kDIM + d] * f;
  }
  out[b * kDIM + d] = ctx / sb;                 // context (L=1,B,DIM)
  if (d == 0) { ws[MS_OFF + b * 2] = mb; ws[MS_OFF + b * 2 + 1] = sb; }
}

// ============ kernel 6: attn[b,t] = exp(E - m_b)/s_b ============
__global__ void k_attn(const float* __restrict__ ws, float* __restrict__ out) {
  const int i = blockIdx.x * 256 + threadIdx.x;  // i = b*T + t
  const int b = i >> 12;
  const float mb = ws[MS_OFF + b * 2], sb = ws[MS_OFF + b * 2 + 1];
  out[kB * kDIM + i] = __expf(ws[E_OFF + i] - mb) / sb;
}

extern "C" void kernel_launch(void* const* d_in, const int* in_sizes, int n_in,
                              void* d_out, int out_size, void* d_ws, size_t ws_size,
                              hipStream_t stream) {
  const float* q     = (const float*)d_in[0];   // (1,B,DIM)
  const float* enc   = (const float*)d_in[1];   // (T,B,DIM)
  const float* mask  = (const float*)d_in[2];   // (B,T,1)
  const float* Ww    = (const float*)d_in[3];   // (DIM,DIM)
  const float* Lw    = (const float*)d_in[4];   // (DIM,NLOC)
  const float* convw = (const float*)d_in[5];   // (NLOC,1,KS)
  const float* convb = (const float*)d_in[6];   // (NLOC,)
  float* out = (float*)d_out;
  float* ws  = (float*)d_ws;

  k_prep<<<1, 256, 0, stream>>>(q, Lw, convw, convb, ws);
  k_wq<<<(kB / 16) * (kDIM / 16), 32, 0, stream>>>(q, Ww, ws + WQ_OFF);
  k_loce<<<(kB * kT) / 256, 256, 0, stream>>>(mask, ws);
  dim3 gmain(kNS, kB);
  k_main<<<gmain, 256, 0, stream>>>(enc, ws);
  k_combine<<<kB, kDIM, 0, stream>>>(ws, out);
  k_attn<<<(kB * kT) / 256, 256, 0, stream>>>(ws, out);
}